// RepulsionLoss_6262062318055
// MI455X (gfx1250) — compile-verified
//
#include <hip/hip_runtime.h>
#include <math.h>

// ---------------------------------------------------------------------------
// RepulsionLoss for MI455X (gfx1250, wave32).
//   dist[b,i,j] = ||p_i||^2 + ||p_j||^2 - 2 p_i.p_j   (clamped at 0)
//   per row: 5 smallest, drop smallest (self), loss += -d * exp(-d/h^2)
//
// Core matrix op: V_WMMA_F32_16X16X4_F32 — K=4 holds the (zero-padded) 3-D
// dot product; one instruction yields a full 16x16 fp32 Gram tile.
// A batch (48 KB) is LDS-resident; HBM traffic is negligible (384 KB total).
// ---------------------------------------------------------------------------

#define NPTS   4096
#define NBATCH 8
#define TILES  (NPTS / 16)        // 256 column tiles
#define H2     (0.03f * 0.03f)    // h^2 = 9e-4

typedef __attribute__((ext_vector_type(2))) float v2f;
typedef __attribute__((ext_vector_type(8))) float v8f;

// Branchless sorted insert: t[0..4] ascending, keep 5 smallest of {t, v}.
__device__ __forceinline__ void insert5(float t[5], float v) {
    float c = v;
#pragma unroll
    for (int k = 0; k < 4; ++k) {
        float lo = fminf(t[k], c);
        c        = fmaxf(t[k], c);
        t[k]     = lo;
    }
    t[4] = fminf(t[4], c);
}

__global__ __launch_bounds__(256)
void repulsion_main(const float* __restrict__ pc, float* __restrict__ partial) {
    __shared__ float s_pc[NPTS * 3];   // 48 KB: whole batch resident in LDS
    __shared__ float s_red[16];

    const int tid      = threadIdx.x;
    const int bid      = blockIdx.x;       // 0..255
    const int batch    = bid >> 5;         // 8 batches, 32 blocks each
    const int blk_in_b = bid & 31;
    const int wave     = tid >> 5;         // 8 waves/block
    const int lane     = tid & 31;
    const int l15      = lane & 15;
    const int hi       = lane >> 4;        // 0: K0/K1 half, 1: K2/K3 half

    // ---- Stage batch point cloud into LDS (float4 vectorized) -------------
    {
        const float4* g = (const float4*)(pc + (size_t)batch * NPTS * 3);
        float4* l = (float4*)s_pc;
        for (int i = tid; i < (NPTS * 3) / 4; i += 256) l[i] = g[i];
    }
    __syncthreads();

    // ---- Row tile for this wave -------------------------------------------
    const int rbase = (blk_in_b * 8 + wave) * 16;
    const float rx = s_pc[(rbase + l15) * 3 + 0];
    const float ry = s_pc[(rbase + l15) * 3 + 1];
    const float rz = s_pc[(rbase + l15) * 3 + 2];
    const float sq_row = rx * rx + ry * ry + rz * rz;

    // A fragment (16x4 fp32, rows pre-scaled by -2):
    //   lanes 0-15:  VGPR0 = K0 = -2x, VGPR1 = K1 = -2y   (M = lane)
    //   lanes 16-31: VGPR0 = K2 = -2z, VGPR1 = K3 = 0     (M = lane-16)
    v2f afrag;
    afrag.x = hi ? (-2.0f * rz) : (-2.0f * rx);
    afrag.y = hi ? 0.0f         : (-2.0f * ry);

    // Row-norm broadcast: accumulator VGPR r holds row (hi*8 + r).
    float crow[8];
#pragma unroll
    for (int r = 0; r < 8; ++r)
        crow[r] = __shfl(sq_row, hi * 8 + r, 32);

    // Per-(lane,row) running 5 smallest distances.
    float t[8][5];
#pragma unroll
    for (int r = 0; r < 8; ++r)
#pragma unroll
        for (int k = 0; k < 5; ++k) t[r][k] = INFINITY;

    // ---- Sweep all 256 column tiles ---------------------------------------
    for (int jt = 0; jt < TILES; ++jt) {
        const int cbase = jt * 16;
        const float cx = s_pc[(cbase + l15) * 3 + 0];
        const float cy = s_pc[(cbase + l15) * 3 + 1];
        const float cz = s_pc[(cbase + l15) * 3 + 2];
        const float sq_col = cx * cx + cy * cy + cz * cz;

        // B fragment (4x16 fp32): mirrors A layout, unscaled.
        v2f bfrag;
        bfrag.x = hi ? cz   : cx;
        bfrag.y = hi ? 0.0f : cy;

        // C = ||row||^2 + ||col||^2 ; D = C - 2*row.col = squared distance.
        v8f c;
#pragma unroll
        for (int r = 0; r < 8; ++r) c[r] = crow[r] + sq_col;

        v8f d = __builtin_amdgcn_wmma_f32_16x16x4_f32(
            false, afrag, false, bfrag, (short)0, c, false, false);

#pragma unroll
        for (int r = 0; r < 8; ++r)
            insert5(t[r], fmaxf(d[r], 0.0f));
    }

    // ---- Merge top-5 lists across the 16 lanes holding each row -----------
#pragma unroll
    for (int s = 1; s < 16; s <<= 1) {
#pragma unroll
        for (int r = 0; r < 8; ++r) {
            float p[5];
#pragma unroll
            for (int k = 0; k < 5; ++k) p[k] = __shfl_xor(t[r][k], s, 32);
#pragma unroll
            for (int k = 0; k < 5; ++k) insert5(t[r], p[k]);
        }
    }

    // ---- Finalize: lane 0 -> rows 0-7, lane 16 -> rows 8-15 ---------------
    float wsum = 0.0f;
    if (l15 == 0) {
#pragma unroll
        for (int r = 0; r < 8; ++r)
#pragma unroll
            for (int k = 1; k < 5; ++k) {      // drop t[r][0] = self
                const float dd = t[r][k];
                wsum += -dd * expf(-dd / H2);
            }
        s_red[wave * 2 + hi] = wsum;
    }
    __syncthreads();

    if (tid == 0) {
        float acc = 0.0f;
#pragma unroll
        for (int i = 0; i < 16; ++i) acc += s_red[i];
        partial[bid] = acc;
    }
}

// Deterministic fixed-order final reduction of 256 block partials.
__global__ __launch_bounds__(256)
void repulsion_reduce(const float* __restrict__ partial, float* __restrict__ out) {
    __shared__ float s[256];
    const int tid = threadIdx.x;
    s[tid] = partial[tid];
    __syncthreads();
    for (int stride = 128; stride > 0; stride >>= 1) {
        if (tid < stride) s[tid] += s[tid + stride];
        __syncthreads();
    }
    if (tid == 0) out[0] = s[0];
}

extern "C" void kernel_launch(void* const* d_in, const int* in_sizes, int n_in,
                              void* d_out, int out_size, void* d_ws, size_t ws_size,
                              hipStream_t stream) {
    const float* pc = (const float*)d_in[0];   // [8, 4096, 3] fp32
    float* out = (float*)d_out;                // scalar fp32 loss
    float* ws  = (float*)d_ws;                 // 256 block partials

    repulsion_main<<<dim3(NBATCH * 32), dim3(256), 0, stream>>>(pc, ws);
    repulsion_reduce<<<dim3(1), dim3(256), 0, stream>>>(ws, out);
}